// SpikeFP32Multiplier_43860206027321
// MI455X (gfx1250) — compile-verified
//
#include <hip/hip_runtime.h>
#include <stdint.h>

// ---------------------------------------------------------------------------
// SpikeFP32Multiplier on MI455X (gfx1250).
// Memory-bound streaming kernel: 805 MB once-through => ~35us floor @23.3TB/s.
// Data path: global_load_async_to_lds_b128 (ASYNCcnt) double-buffered staging,
// ds_load_b128 consume, integer bit pack/unpack, one v_mul_f32 (denorms on,
// RNE by default), non-temporal global_store_b128 output.
// ---------------------------------------------------------------------------

typedef __attribute__((ext_vector_type(4))) unsigned int v4u;
typedef int vi4 __attribute__((vector_size(16)));  // matches builtin param type

#define TILE 64   // elements per tile == threads per block (2 wave32 waves)
#define VPE  8    // 16-byte vectors per element (32 floats = 128 B)

#if defined(__has_builtin)
#if __has_builtin(__builtin_amdgcn_global_load_async_to_lds_b128)
#define HAS_ASYNC_B128 1
#endif
#if __has_builtin(__builtin_amdgcn_s_wait_asynccnt)
#define HAS_WAIT_ASYNC 1
#endif
#endif

typedef __attribute__((address_space(1))) vi4 gvi4;  // global int4
typedef __attribute__((address_space(3))) vi4 lvi4;  // LDS int4

__device__ __forceinline__ void async_copy_b128(const v4u* g, v4u* l) {
#ifdef HAS_ASYNC_B128
  __builtin_amdgcn_global_load_async_to_lds_b128((gvi4*)g, (lvi4*)l,
                                                 /*offset=*/0, /*cpol=*/0);
#else
  // GV mode: vdst = LDS byte offset, vaddr = 64-bit global address.
  asm volatile("global_load_async_to_lds_b128 %0, %1, off"
               :
               : "v"((unsigned)(uintptr_t)l), "v"(g)
               : "memory");
#endif
}

__device__ __forceinline__ void wait_async_le16() {
#ifdef HAS_WAIT_ASYNC
  __builtin_amdgcn_s_wait_asynccnt(16);
#else
  asm volatile("s_wait_asynccnt 0x10" ::: "memory");
#endif
}

__device__ __forceinline__ void wait_async_0() {
#ifdef HAS_WAIT_ASYNC
  __builtin_amdgcn_s_wait_asynccnt(0);
#else
  asm volatile("s_wait_asynccnt 0x0" ::: "memory");
#endif
}

// Inputs are exactly 0.0f (0x00000000) or 1.0f (0x3F800000):
// spike bit == bit 23 of the raw word. Spike index j maps to IEEE bit (31-j).
__device__ __forceinline__ void pack4(v4u q, int k, unsigned& w) {
  w |= ((q.x >> 23) & 1u) << (31 - (4 * k + 0));
  w |= ((q.y >> 23) & 1u) << (31 - (4 * k + 1));
  w |= ((q.z >> 23) & 1u) << (31 - (4 * k + 2));
  w |= ((q.w >> 23) & 1u) << (31 - (4 * k + 3));
}

__device__ __forceinline__ unsigned pack_word(const v4u* l) {
  unsigned w = 0;
#pragma unroll
  for (int k = 0; k < VPE; ++k) pack4(l[k], k, w);
  return w;
}

__device__ __forceinline__ void unpack_store(unsigned wp, v4u* o) {
#pragma unroll
  for (int k = 0; k < VPE; ++k) {
    v4u q;
    q.x = ((wp >> (31 - (4 * k + 0))) & 1u) ? 0x3F800000u : 0u;
    q.y = ((wp >> (31 - (4 * k + 1))) & 1u) ? 0x3F800000u : 0u;
    q.z = ((wp >> (31 - (4 * k + 2))) & 1u) ? 0x3F800000u : 0u;
    q.w = ((wp >> (31 - (4 * k + 3))) & 1u) ? 0x3F800000u : 0u;
    __builtin_nontemporal_store(q, o + k);  // write-once stream: th=NT
  }
}

// Stage one tile (TILE elements of A and B) into an LDS buffer: 16 async b128
// ops per thread. Thread-private LDS regions -> no workgroup barrier needed.
__device__ __forceinline__ void stage_tile(const v4u* A, const v4u* B, int t,
                                           int tid, v4u* dst) {
  const size_t e = (size_t)t * TILE + tid;
  const v4u* ga = A + e * VPE;
  const v4u* gb = B + e * VPE;
  v4u* la = dst + (0 * TILE + tid) * VPE;
  v4u* lb = dst + (1 * TILE + tid) * VPE;
#pragma unroll
  for (int k = 0; k < VPE; ++k) async_copy_b128(ga + k, la + k);
#pragma unroll
  for (int k = 0; k < VPE; ++k) async_copy_b128(gb + k, lb + k);
}

__global__ __launch_bounds__(TILE) void spike_fp32_mul_async(
    const float* __restrict__ Af, const float* __restrict__ Bf,
    float* __restrict__ Of, int n_tiles) {
  // [2 buffers][{A,B}][TILE elems][8 x 16B] = 32 KB
  __shared__ v4u lds[2 * 2 * TILE * VPE];
  const v4u* A = (const v4u*)Af;
  const v4u* B = (const v4u*)Bf;
  v4u* O = (v4u*)Of;
  const int tid = threadIdx.x;
  const int G = gridDim.x;

  int t = blockIdx.x;
  if (t >= n_tiles) return;

  stage_tile(A, B, t, tid, &lds[0]);  // prologue: 16 async ops in flight
  int buf = 0;
  for (;;) {
    const int tn = t + G;                 // uniform across the block
    const bool have_next = (tn < n_tiles);
    if (have_next)
      stage_tile(A, B, tn, tid, &lds[(buf ^ 1) * 2 * TILE * VPE]);
    // Async loads complete in order: <=16 outstanding => current tile is
    // resident in LDS; ==0 on the drain iteration.
    if (have_next) wait_async_le16(); else wait_async_0();
    asm volatile("" ::: "memory");        // keep ds_loads below the wait

    const v4u* la = &lds[(buf * 2 + 0) * TILE * VPE + tid * VPE];
    const v4u* lb = &lds[(buf * 2 + 1) * TILE * VPE + tid * VPE];
    const unsigned wa = pack_word(la);    // ds_load_b128 x8 + bit pack
    const unsigned wb = pack_word(lb);
    const float fp = __uint_as_float(wa) * __uint_as_float(wb);  // IEEE RNE,
    const unsigned wp = __float_as_uint(fp);                     // denorms on
    unpack_store(wp, O + ((size_t)t * TILE + tid) * VPE);

    if (!have_next) break;
    t = tn;
    buf ^= 1;
  }
}

// Tail path (direct global b128 loads) for element counts not divisible by
// TILE. Not taken for the harness shape (2^21 elements), but keeps the
// kernel shape-general.
__global__ void spike_fp32_mul_tail(const float* __restrict__ Af,
                                    const float* __restrict__ Bf,
                                    float* __restrict__ Of, long long start,
                                    long long n_elems) {
  long long e = start + (long long)blockIdx.x * blockDim.x + threadIdx.x;
  if (e >= n_elems) return;
  const v4u* ga = (const v4u*)Af + (size_t)e * VPE;
  const v4u* gb = (const v4u*)Bf + (size_t)e * VPE;
  unsigned wa = 0, wb = 0;
#pragma unroll
  for (int k = 0; k < VPE; ++k) {
    pack4(ga[k], k, wa);
    pack4(gb[k], k, wb);
  }
  const float fp = __uint_as_float(wa) * __uint_as_float(wb);
  unpack_store(__float_as_uint(fp), (v4u*)Of + (size_t)e * VPE);
}

extern "C" void kernel_launch(void* const* d_in, const int* in_sizes, int n_in,
                              void* d_out, int out_size, void* d_ws,
                              size_t ws_size, hipStream_t stream) {
  (void)n_in; (void)out_size; (void)d_ws; (void)ws_size;
  const float* A = (const float*)d_in[0];
  const float* B = (const float*)d_in[1];
  float* O = (float*)d_out;

  const long long n_elems = (long long)in_sizes[0] / 32;  // 32 spike bits each
  const int n_tiles = (int)(n_elems / TILE);
  const long long tail_start = (long long)n_tiles * TILE;

  if (n_tiles > 0) {
    // Persistent-ish grid: >=4 tiles per block so double buffering pays off;
    // 4096 blocks x 2 waves saturates every WGP several times over.
    const int blocks = n_tiles < 4096 ? n_tiles : 4096;
    spike_fp32_mul_async<<<blocks, TILE, 0, stream>>>(A, B, O, n_tiles);
  }
  if (tail_start < n_elems) {
    const int rem = (int)(n_elems - tail_start);
    spike_fp32_mul_tail<<<(rem + TILE - 1) / TILE, TILE, 0, stream>>>(
        A, B, O, tail_start, n_elems);
  }
}